// HierarchialModel_86569360818845
// MI455X (gfx1250) — compile-verified
//
#include <hip/hip_runtime.h>

typedef __attribute__((ext_vector_type(2))) float v2f;
typedef __attribute__((ext_vector_type(8))) float v8f;

// One wave32 per batch element.
// Tile = 16 tree levels -> A matrix (16 x 128 fp32), consumed K=4 per
// V_WMMA_F32_16X16X4_F32; B matrix = h broadcast over all 16 N columns.
__global__ __launch_bounds__(256)
void hs_path_prob_kernel(const float* __restrict__ encoder,
                         const float* __restrict__ logistic,
                         const int*   __restrict__ v_j,
                         const int*   __restrict__ path_idx,
                         const int*   __restrict__ path_sign,
                         float*       __restrict__ out,
                         int B, int L, int numTiles)
{
    const int tid  = blockIdx.x * blockDim.x + threadIdx.x;
    const int wave = tid >> 5;            // batch element
    const int lane = threadIdx.x & 31;
    if (wave >= B) return;

    const int half = lane >> 4;           // 0 -> K pair {0,1}; 1 -> K pair {2,3}
    const int m    = lane & 15;           // A-matrix row (level within tile)

    const int vj = v_j[wave];
    const float* hptr = encoder + (size_t)vj * 128 + half * 2;

    // h chunks, reused by every tile.  hb[i] supplies B-matrix rows
    // K = {4i + 2*half, 4i + 2*half + 1}, identical across N (broadcast).
    v2f hb[32];
#pragma unroll
    for (int i = 0; i < 32; ++i)
        hb[i] = *(const v2f*)(hptr + 4 * i);

    float prod = 1.0f;

    for (int tile = 0; tile < numTiles; ++tile) {
        const int lev = tile * 16 + m;
        int row = 0, sgn = 0;             // sgn==0 <=> masked-out (prob = 1)
        if (lev < L) {
            row = path_idx[(size_t)wave * L + lev];
            sgn = path_sign[(size_t)wave * L + lev];
        }
        const float* wptr = logistic + (size_t)row * 128 + half * 2;

        // Bulk gather this tile's 16 logistic rows (256B per lane as 32x b64).
        v2f wa[32];
#pragma unroll
        for (int i = 0; i < 32; ++i)
            wa[i] = *(const v2f*)(wptr + 4 * i);

        // D = sum_k A[m,k]*h[k], accumulated over 32 K-chunks of 4.
        v8f c = {0.f, 0.f, 0.f, 0.f, 0.f, 0.f, 0.f, 0.f};
#pragma unroll
        for (int i = 0; i < 32; ++i)
            c = __builtin_amdgcn_wmma_f32_16x16x4_f32(
                    false, wa[i], false, hb[i], (short)0, c, false, false);

        // C/D layout: VGPR v on lanes 0-15 holds level tile*16+v,
        // on lanes 16-31 holds level tile*16+v+8 (all N columns equal).
        const int base8 = half * 8;
        float p = 1.0f;
#pragma unroll
        for (int j = 0; j < 8; ++j) {
            int s = __shfl(sgn, base8 + j, 32);  // sign of level base8+j (lane base8+j holds it)
            if (s != 0) {
                float x = c[j] * (float)s;
                p *= 1.0f / (1.0f + __expf(-x));
            }
        }
        p *= __shfl_xor(p, 16, 32);       // combine the two half-wave products
        prod *= p;
    }

    if (lane == 0) out[wave] = prod;
}

extern "C" void kernel_launch(void* const* d_in, const int* in_sizes, int n_in,
                              void* d_out, int out_size, void* d_ws, size_t ws_size,
                              hipStream_t stream)
{
    const float* encoder   = (const float*)d_in[0];
    const float* logistic  = (const float*)d_in[1];
    const int*   v_j       = (const int*)  d_in[2];
    const int*   path_idx  = (const int*)  d_in[3];
    const int*   path_sign = (const int*)  d_in[4];
    // d_in[5] = path_mask: unused -- path_sign==0 encodes mask==false.
    float* out = (float*)d_out;

    const int B = in_sizes[2];            // 8192 pairs
    const int L = in_sizes[3] / B;        // padded path length (~19)
    const int numTiles = (L + 15) / 16;   // 16 levels per WMMA tile

    const int threads = 256;              // 8 waves -> 8 batch elems per block
    const int wavesPerBlock = threads / 32;
    const int blocks = (B + wavesPerBlock - 1) / wavesPerBlock;

    hipLaunchKernelGGL(hs_path_prob_kernel, dim3(blocks), dim3(threads), 0, stream,
                       encoder, logistic, v_j, path_idx, path_sign, out,
                       B, L, numTiles);
}